// SoftMLoss_14929306321138
// MI455X (gfx1250) — compile-verified
//
#include <hip/hip_runtime.h>
#include <math.h>

// ---------------------------------------------------------------------------
// SoftMLoss on MI455X (gfx1250, wave32).
//   dist = euclidean_distance_matrix(x)         via fp32 WMMA (16x16x4)
//   loss = mean over 14336 (row, pos) pairs of
//          log(S_out(row) + exp(d[row,pos]-M(row))) + M(row) - d[row,pos]
// where S_out(row) sums exp(d-M) over all columns outside row's 8-block.
// ---------------------------------------------------------------------------

#define BATCH 4096
#define DIM   256
#define FPC   8
#define NBLK  (BATCH / FPC)   // 512
#define NPAIRS (NBLK * 28)    // 14336

typedef __attribute__((ext_vector_type(2))) float v2f;
typedef __attribute__((ext_vector_type(8))) float v8f;

// ---------------- K0: squared row norms -----------------------------------
__global__ __launch_bounds__(256) void sqnorm_kernel(const float* __restrict__ x,
                                                     float* __restrict__ sq) {
  __shared__ float red[256];
  const int r = blockIdx.x;
  const int t = threadIdx.x;
  const float v = x[(size_t)r * DIM + t];
  red[t] = v * v;
  __syncthreads();
  #pragma unroll
  for (int s = 128; s > 0; s >>= 1) {
    if (t < s) red[t] += red[t + s];
    __syncthreads();
  }
  if (t == 0) sq[r] = red[0];
}

// ---------------- K1: Gram via WMMA + distance epilogue -------------------
#define BM 128
#define BN 128
#define BK 32
#define LDA (BK + 2)  // padded LDS row stride (floats), keeps banks spread

__global__ __launch_bounds__(256) void gemm_dist_kernel(const float* __restrict__ x,
                                                        const float* __restrict__ sq,
                                                        float* __restrict__ dist) {
  __shared__ float lA[BM * LDA];
  __shared__ float lB[BN * LDA];
  const int tid   = threadIdx.x;
  const int lane  = tid & 31;
  const int wid   = tid >> 5;     // 8 waves
  const int waveM = wid & 3;      // 4 row stripes of 32
  const int waveN = wid >> 2;     // 2 col stripes of 64
  const int row0  = blockIdx.y * BM;
  const int col0  = blockIdx.x * BN;

  v8f acc[2][4];
  #pragma unroll
  for (int mt = 0; mt < 2; ++mt)
    #pragma unroll
    for (int nt = 0; nt < 4; ++nt)
      acc[mt][nt] = (v8f){};

  const int mr   = lane & 15;          // row within 16-tile (A) / col (B)
  const int hi   = lane >> 4;          // lane-half selects K offset {0,2}
  const int koff = hi << 1;

  for (int k0 = 0; k0 < DIM; k0 += BK) {
    // Cooperative tile load: 128x32 floats each for A and B rows of X.
    #pragma unroll
    for (int it = 0; it < 4; ++it) {
      const int idx = tid + it * 256;    // float4 index 0..1023
      const int rr  = idx >> 3;          // 8 float4 per row
      const int cc  = (idx & 7) << 2;
      const float4 a4 = *(const float4*)(x + (size_t)(row0 + rr) * DIM + k0 + cc);
      const float4 b4 = *(const float4*)(x + (size_t)(col0 + rr) * DIM + k0 + cc);
      float* pa = &lA[rr * LDA + cc];
      pa[0] = a4.x; pa[1] = a4.y; pa[2] = a4.z; pa[3] = a4.w;
      float* pb = &lB[rr * LDA + cc];
      pb[0] = b4.x; pb[1] = b4.y; pb[2] = b4.z; pb[3] = b4.w;
    }
    // Prefetch next k-tile (global_prefetch_b8).
    if (k0 + BK < DIM) {
      const int rr = tid & 127;
      const float* p = (tid < 128 ? x + (size_t)(row0 + rr) * DIM
                                  : x + (size_t)(col0 + rr) * DIM) + (k0 + BK);
      __builtin_prefetch(p, 0, 0);
    }
    __syncthreads();

    #pragma unroll
    for (int kk = 0; kk < BK; kk += 4) {
      v2f afrag[2], bfrag[4];
      #pragma unroll
      for (int mt = 0; mt < 2; ++mt) {
        const float* p = &lA[(waveM * 32 + mt * 16 + mr) * LDA + kk + koff];
        v2f a; a.x = p[0]; a.y = p[1];
        afrag[mt] = a;
      }
      #pragma unroll
      for (int nt = 0; nt < 4; ++nt) {
        const float* p = &lB[(waveN * 64 + nt * 16 + mr) * LDA + kk + koff];
        v2f b; b.x = p[0]; b.y = p[1];
        bfrag[nt] = b;
      }
      #pragma unroll
      for (int mt = 0; mt < 2; ++mt)
        #pragma unroll
        for (int nt = 0; nt < 4; ++nt)
          acc[mt][nt] = __builtin_amdgcn_wmma_f32_16x16x4_f32(
              false, afrag[mt], false, bfrag[nt],
              (short)0, acc[mt][nt], false, false);
    }
    __syncthreads();
  }

  // Epilogue: d2 = sq[r] + sq[c] - 2*dot; dist = (d2==0) ? 0 : sqrt(d2)
  #pragma unroll
  for (int nt = 0; nt < 4; ++nt) {
    const int col = col0 + waveN * 64 + nt * 16 + mr;
    const float sqc = sq[col];
    #pragma unroll
    for (int mt = 0; mt < 2; ++mt) {
      #pragma unroll
      for (int v = 0; v < 8; ++v) {
        const int row = row0 + waveM * 32 + mt * 16 + hi * 8 + v;
        float d2 = sq[row] + sqc - 2.0f * acc[mt][nt][v];
        d2 = fmaxf(d2, 0.0f);
        const float d = (d2 == 0.0f) ? 0.0f : sqrtf(d2);
        dist[(size_t)row * BATCH + col] = d;
      }
    }
  }
}

// ---------------- K2: per-row max + out-of-block exp sum ------------------
__global__ __launch_bounds__(256) void rowstat_kernel(const float* __restrict__ dist,
                                                      float* __restrict__ rmax,
                                                      float* __restrict__ rsum) {
  __shared__ float red[256];
  const int r = blockIdx.x;
  const int t = threadIdx.x;
  const int blk0 = (r / FPC) * FPC;
  const float* __restrict__ drow = dist + (size_t)r * BATCH;

  float m = -3.0e38f;
  for (int j = t; j < BATCH; j += 256) m = fmaxf(m, drow[j]);
  red[t] = m;
  __syncthreads();
  #pragma unroll
  for (int s = 128; s > 0; s >>= 1) {
    if (t < s) red[t] = fmaxf(red[t], red[t + s]);
    __syncthreads();
  }
  const float M = red[0];
  __syncthreads();

  float sum = 0.0f;
  for (int j = t; j < BATCH; j += 256) {
    const float e = expf(drow[j] - M);
    const bool inblk = (j >= blk0) && (j < blk0 + FPC);
    sum += inblk ? 0.0f : e;
  }
  red[t] = sum;
  __syncthreads();
  #pragma unroll
  for (int s = 128; s > 0; s >>= 1) {
    if (t < s) red[t] += red[t + s];
    __syncthreads();
  }
  if (t == 0) { rmax[r] = M; rsum[r] = red[0]; }
}

// ---------------- K3: per-pair NLL, one wave per 8-block ------------------
__global__ __launch_bounds__(32) void pair_kernel(const float* __restrict__ dist,
                                                  const float* __restrict__ rmax,
                                                  const float* __restrict__ rsum,
                                                  float* __restrict__ partial) {
  const int blk = blockIdx.x;
  const int t = threadIdx.x;
  float nll = 0.0f;
  if (t < 28) {
    int a = 0, p = t, cnt = 7;           // decode pair index -> (a,b), a<b<8
    while (p >= cnt) { p -= cnt; --cnt; ++a; }
    const int b = a + 1 + p;
    const int row = blk * FPC + a;
    const int pos = blk * FPC + b;
    const float d = dist[(size_t)row * BATCH + pos];
    const float M = rmax[row];
    const float S = rsum[row];
    nll = logf(S + expf(d - M)) + M - d;
  }
  #pragma unroll
  for (int off = 16; off > 0; off >>= 1) nll += __shfl_down(nll, off);
  if (t == 0) partial[blk] = nll;
}

// ---------------- K4: final mean ------------------------------------------
__global__ __launch_bounds__(256) void final_kernel(const float* __restrict__ partial,
                                                    float* __restrict__ out) {
  __shared__ float red[256];
  const int t = threadIdx.x;
  float s = 0.0f;
  for (int i = t; i < NBLK; i += 256) s += partial[i];
  red[t] = s;
  __syncthreads();
  #pragma unroll
  for (int st = 128; st > 0; st >>= 1) {
    if (t < st) red[t] += red[t + st];
    __syncthreads();
  }
  if (t == 0) out[0] = red[0] / (float)NPAIRS;
}

// ---------------------------------------------------------------------------
extern "C" void kernel_launch(void* const* d_in, const int* in_sizes, int n_in,
                              void* d_out, int out_size, void* d_ws, size_t ws_size,
                              hipStream_t stream) {
  const float* x = (const float*)d_in[0];
  float* out = (float*)d_out;

  char* ws = (char*)d_ws;
  float* dist    = (float*)ws;                              // 4096*4096 f32 = 64 MB
  float* sq      = (float*)(ws + (size_t)BATCH * BATCH * sizeof(float));
  float* rmax    = sq + BATCH;
  float* rsum    = rmax + BATCH;
  float* partial = rsum + BATCH;                            // NBLK floats

  sqnorm_kernel<<<BATCH, 256, 0, stream>>>(x, sq);
  gemm_dist_kernel<<<dim3(BATCH / BN, BATCH / BM), 256, 0, stream>>>(x, sq, dist);
  rowstat_kernel<<<BATCH, 256, 0, stream>>>(dist, rmax, rsum);
  pair_kernel<<<NBLK, 32, 0, stream>>>(dist, rmax, rsum, partial);
  final_kernel<<<1, 256, 0, stream>>>(partial, out);
}